// WaveletTransformLayer_86955907874874
// MI455X (gfx1250) — compile-verified
//
#include <hip/hip_runtime.h>

// Haar wavelet butterfly along S for x:(B=16, S=8192, C=512) f32.
//   cA[b,s,:] = (x[b,2s,:] + x[b,2s+1,:]) * 1/sqrt(2)   -> out rows [0,    4096)
//   cD[b,s,:] = (x[b,2s,:] - x[b,2s+1,:]) * 1/sqrt(2)   -> out rows [4096, 8192)
//
// Memory-bound (0.25 FLOP/byte): pure 128-bit NT streaming kernel.
// 8 pair-rows per block to amortize workgroup setup / wave launch and to put
// 16 independent b128 NT loads in flight per thread before the first wait.

typedef __attribute__((ext_vector_type(4))) float v4f;

#define HAAR_B 16
#define HAAR_S 8192
#define HAAR_C 512
#define HAAR_CQ (HAAR_C / 4)          // 128 float4 per channel row
#define HAAR_HALF (HAAR_S / 2)        // 4096
#define ROWS_PER_BLOCK 8
#define INV_SQRT2F 0.70710678118654752440f

__global__ __launch_bounds__(HAAR_CQ) void haar_fwd_kernel(
    const float* __restrict__ x, float* __restrict__ out) {
  const unsigned c4 = threadIdx.x;                 // channel quad: 0..127
  const unsigned s0 = blockIdx.x * ROWS_PER_BLOCK; // first output row of block
  const unsigned b  = blockIdx.y;                  // batch: 0..15

  // Input rows 2*s0 .. 2*s0+15; each row is HAAR_CQ float4 wide.
  const v4f* pin = (const v4f*)x +
      (((size_t)b * HAAR_S + 2u * (size_t)s0) * HAAR_CQ + c4);

  v4f e[ROWS_PER_BLOCK], o[ROWS_PER_BLOCK];
#pragma unroll
  for (int r = 0; r < ROWS_PER_BLOCK; ++r) {
    e[r] = __builtin_nontemporal_load(pin + (2 * r) * HAAR_CQ);      // row 2(s0+r)
    o[r] = __builtin_nontemporal_load(pin + (2 * r + 1) * HAAR_CQ);  // row 2(s0+r)+1
  }

  v4f* pa = (v4f*)out + (((size_t)b * HAAR_S + (size_t)s0) * HAAR_CQ + c4);
  v4f* pd = pa + (size_t)HAAR_HALF * HAAR_CQ;

#pragma unroll
  for (int r = 0; r < ROWS_PER_BLOCK; ++r) {
    const v4f a = (e[r] + o[r]) * INV_SQRT2F;
    const v4f d = (e[r] - o[r]) * INV_SQRT2F;
    __builtin_nontemporal_store(a, pa + r * HAAR_CQ);
    __builtin_nontemporal_store(d, pd + r * HAAR_CQ);
  }
}

extern "C" void kernel_launch(void* const* d_in, const int* in_sizes, int n_in,
                              void* d_out, int out_size, void* d_ws, size_t ws_size,
                              hipStream_t stream) {
  (void)n_in; (void)out_size; (void)d_ws; (void)ws_size;
  const float* x = (const float*)d_in[0];
  float* out = (float*)d_out;

  // Derive batch count from the flat input size (S, C fixed by the reference).
  const int b = in_sizes[0] / (HAAR_S * HAAR_C);

  dim3 grid(HAAR_HALF / ROWS_PER_BLOCK, (unsigned)b, 1);
  dim3 block(HAAR_CQ, 1, 1);
  haar_fwd_kernel<<<grid, block, 0, stream>>>(x, out);
}